// HMC_50964081934395
// MI455X (gfx1250) — compile-verified
//
#include <hip/hip_runtime.h>

typedef __attribute__((ext_vector_type(2))) float v2f;
typedef __attribute__((ext_vector_type(8))) float v8f;
typedef __attribute__((ext_vector_type(4))) unsigned int v4u;
typedef __attribute__((ext_vector_type(4))) int v4i;
typedef __attribute__((ext_vector_type(8))) int v8i;

#if defined(__HIP_DEVICE_COMPILE__) && !__has_builtin(__builtin_amdgcn_wmma_f32_16x16x4_f32)
#error "missing __builtin_amdgcn_wmma_f32_16x16x4_f32 on this target"
#endif

#if __has_builtin(__builtin_amdgcn_tensor_load_to_lds) && __has_builtin(__builtin_amdgcn_s_wait_tensorcnt)
#define USE_TDM 1
#else
#define USE_TDM 0
#endif

#define T_STEPS  30
#define L_STEPS  5
#define DIM      128
#define BATCH    512
#define CHAINS   32
#define RPB      32                       // chains per block (== CHAINS -> block b handles batch b)
#define ZS       132                      // padded LDS row stride (floats): bank-conflict-free frags
#define AST      132                      // padded LDS row stride for A
#define NTHREADS 256                      // 8 wave32
#define EPB      ((RPB * DIM) / NTHREADS) // 16 elements owned per thread

#define FLT_MAX_ 3.402823466e+38f

// ---- dynamic LDS layout (in floats) ----
#define OFF_A    0
#define OFF_Z    (OFF_A   + DIM * AST)
#define OFF_ZSV  (OFF_Z   + RPB * ZS)
#define OFF_G    (OFF_ZSV + RPB * ZS)
#define OFF_R0   (OFF_G   + RPB * ZS)
#define OFF_R1   (OFF_R0  + RPB * ZS)
#define OFF_EPS  (OFF_R1  + RPB * ZS)
#define OFF_MU   (OFF_EPS + DIM)
#define OFF_SB   (OFF_MU  + DIM)
#define OFF_ACC  (OFF_SB  + DIM)
#define OFF_ILL  (OFF_ACC + RPB)
#define SMEM_FLOATS (OFF_ILL + 1)

// ---------------------------------------------------------------------------
// TDM: async DMA of one step's momentum noise (RPB x DIM f32, row-contiguous
// in global) into LDS with hardware padding to the ZS=132 layout.
// D# per CDNA5 ISA:
//   group0: count=1 | lds_addr | global_addr[56:0] | type=2
//   group1: data_size=2(4B), pad_enable=1, pad_interval=6 (128 DWORDs),
//           pad_amount=3 (4 DWORDs), tensor_dim0=128, tensor_dim1=32,
//           tile_dim0=128, tile_dim1=32, tensor_dim0_stride=128
// This toolchain exposes the 6-arg builtin:
//   (uint32x4 g0, int32x8 g1, int32x4 g2, int32x4 g3, int32x8, i32 cpol)
// ---------------------------------------------------------------------------
#if USE_TDM
__device__ __forceinline__ void tdm_load_r(const float* gsrc, unsigned lds_byte)
{
    unsigned long long ga = (unsigned long long)(uintptr_t)gsrc;
    v4u g0;
    g0.x = 1u;                                                  // count=1, user mode
    g0.y = lds_byte;                                            // lds_addr
    g0.z = (unsigned)ga;                                        // global_addr lo
    g0.w = (unsigned)((ga >> 32) & 0x1FFFFFFu) | (2u << 30);    // global_addr hi | type=2
    v8i g1;
    g1[0] = (int)((2u << 16) | (1u << 20) | (6u << 22) | (3u << 25));
    g1[1] = (int)(128u << 16);   // tensor_dim0[15:0] in [31:16]
    g1[2] = (int)(32u << 16);    // tensor_dim0 hi=0 ; tensor_dim1[15:0]=32
    g1[3] = (int)(128u << 16);   // tensor_dim1 hi=0 ; tile_dim0=128
    g1[4] = 32;                  // tile_dim1=32 ; tile_dim2=0
    g1[5] = 128;                 // tensor_dim0_stride lo32
    g1[6] = 0;                   // stride0 hi ; stride1 lo
    g1[7] = 0;                   // stride1 hi
    v4i z4 = {0, 0, 0, 0};
    v8i z8 = {0, 0, 0, 0, 0, 0, 0, 0};
    __builtin_amdgcn_tensor_load_to_lds(g0, g1, z4, z4, z8, 0);
}
#endif

// G = z @ A   (z: RPB x DIM in LDS stride ZS; A: DIM x DIM in LDS stride AST)
// wave w: rows m0 = (w>>2)*16, cols nb = (w&3)*32 (two 16-wide N tiles).
// V_WMMA_F32_16X16X4_F32 fragment layouts per CDNA5 ISA 7.12.2.
__device__ __forceinline__ void gemm_zA(const float* __restrict__ zl,
                                        const float* __restrict__ Al,
                                        float* __restrict__ gl, int tid)
{
    const int lane = tid & 31;
    const int li   = lane & 15;
    const int half = lane >> 4;
    const int wv   = tid >> 5;
    const int m0   = (wv >> 2) << 4;   // 0 or 16
    const int nb   = (wv & 3) << 5;    // 0,32,64,96

    v8f accA0 = {}, accA1 = {}, accB0 = {}, accB1 = {};  // 4 independent chains
    const float* zrow = zl + (m0 + li) * ZS + 2 * half;

#pragma unroll
    for (int k0 = 0; k0 < DIM; k0 += 8) {
        v2f a0 = *(const v2f*)(zrow + k0);
        v2f a1 = *(const v2f*)(zrow + k0 + 4);
        const float* Ar0 = Al + (k0 + 2 * half) * AST + nb + li;
        const float* Ar1 = Ar0 + 4 * AST;
        v2f b00 = { Ar0[0],  Ar0[AST]      };
        v2f b01 = { Ar0[16], Ar0[AST + 16] };
        v2f b10 = { Ar1[0],  Ar1[AST]      };
        v2f b11 = { Ar1[16], Ar1[AST + 16] };
        accA0 = __builtin_amdgcn_wmma_f32_16x16x4_f32(false, a0, false, b00, (short)0, accA0, false, false);
        accA1 = __builtin_amdgcn_wmma_f32_16x16x4_f32(false, a0, false, b01, (short)0, accA1, false, false);
        accB0 = __builtin_amdgcn_wmma_f32_16x16x4_f32(false, a1, false, b10, (short)0, accB0, false, false);
        accB1 = __builtin_amdgcn_wmma_f32_16x16x4_f32(false, a1, false, b11, (short)0, accB1, false, false);
    }
    v8f acc0 = accA0 + accB0;
    v8f acc1 = accA1 + accB1;

    float* grow = gl + (m0 + 8 * half) * ZS + nb + li;
#pragma unroll
    for (int v = 0; v < 8; ++v) {
        grow[v * ZS]      = acc0[v];
        grow[v * ZS + 16] = acc1[v];
    }
}

__global__ __launch_bounds__(NTHREADS, 1)
void hmc_persistent_kernel(const float* __restrict__ mu0,
                           const float* __restrict__ var0,
                           const float* __restrict__ log_eps,
                           const float* __restrict__ log_inflation,
                           const float* __restrict__ Ag,
                           const float* __restrict__ noise_init,
                           const float* __restrict__ noise_r,
                           const float* __restrict__ u_accept,
                           float* __restrict__ out)
{
    extern __shared__ float smem[];
    float* Al    = smem + OFF_A;
    float* zl    = smem + OFF_Z;
    float* zsv   = smem + OFF_ZSV;
    float* gl    = smem + OFF_G;
    float* rb0   = smem + OFF_R0;
    float* rb1   = smem + OFF_R1;
    float* epsl  = smem + OFF_EPS;
    float* muL   = smem + OFF_MU;
    float* sbL   = smem + OFF_SB;
    float* accl  = smem + OFF_ACC;
    int*   anyill = (int*)(smem + OFF_ILL);

    const int tid  = threadIdx.x;
    const int b    = blockIdx.x;         // one block per batch element
    const int row0 = b * CHAINS;         // first global chain row
    const size_t stride_t = (size_t)BATCH * CHAINS * DIM;

    float* z_final = out;
    float* z_stack = out + stride_t;     // 31 slots

#if USE_TDM
    // kick off async DMA of step-0 momentum noise into r buffer 0 (wave 0 only)
    if (tid < 32)
        tdm_load_r(noise_r + (size_t)row0 * DIM, (unsigned)(uintptr_t)rb0);
#endif

    // ---- setup: A into LDS (padded), per-batch mu / (sigma * inflation) ----
#pragma unroll
    for (int i = 0; i < (DIM * DIM) / NTHREADS; ++i) {
        int idx = i * NTHREADS + tid;
        Al[(idx >> 7) * AST + (idx & 127)] = Ag[idx];
    }
    if (tid < DIM) {
        float infl = expf(log_inflation[0]);
        muL[tid] = mu0[b * DIM + tid];
        sbL[tid] = sqrtf(var0[b * DIM + tid]) * infl;
    }
    __syncthreads();

    // ---- z0 = noise_init * (sigma*inflation) + mu ; emit z_stack[0] ----
#pragma unroll
    for (int i = 0; i < EPB; ++i) {
        int e = i * NTHREADS + tid;
        int rr = e >> 7, d = e & 127;
        float zv = noise_init[(size_t)row0 * DIM + e] * sbL[d] + muL[d];
        zl[rr * ZS + d] = zv;
        z_stack[(size_t)row0 * DIM + e] = zv;
    }
    __syncthreads();

    for (int t = 0; t < T_STEPS; ++t) {
        float* rl = (t & 1) ? rb1 : rb0;
        const size_t nbase = (size_t)t * stride_t + (size_t)row0 * DIM;

        // ---- phase A: eps, save z; momentum arrives via TDM (log_v_r==0 -> r=noise) ----
        if (tid < DIM) epsl[tid] = expf(log_eps[t * DIM + tid]);
        if (tid == 0) *anyill = 0;
        unsigned illm = 0;
#pragma unroll
        for (int i = 0; i < EPB; ++i) {
            int e = i * NTHREADS + tid;
            int o = (e >> 7) * ZS + (e & 127);
#if !USE_TDM
            rl[o] = noise_r[nbase + e];
#endif
            zsv[o] = zl[o];
        }
#if USE_TDM
        if (tid < 32) __builtin_amdgcn_s_wait_tensorcnt((short)0);  // r for step t landed
#endif
        __syncthreads();

#if USE_TDM
        // prefetch next step's momentum into the other buffer; runs under the leapfrog
        if (t + 1 < T_STEPS && tid < 32)
            tdm_load_r(noise_r + nbase + stride_t,
                       (unsigned)(uintptr_t)((t & 1) ? rb0 : rb1));
#endif

        // ---- phase B: initial kinetic energy (wave0, lane==chain) + GEMM g = z@A ----
        float H0 = 0.f;
        if (tid < RPB) {
            float kin = 0.f;
            for (int d = 0; d < DIM; ++d) { float rv = rl[tid * ZS + d]; kin += rv * rv; }
            H0 = 0.5f * kin;
        }
        gemm_zA(zl, Al, gl, tid);
        __syncthreads();

        // ---- phase C: pot_init = 0.5 * <g,z> (reuses GEMM) ; half kick ----
        if (tid < RPB) {
            float pot = 0.f;
            for (int d = 0; d < DIM; ++d) pot += gl[tid * ZS + d] * zl[tid * ZS + d];
            H0 += 0.5f * pot;
        }
#pragma unroll
        for (int i = 0; i < EPB; ++i) {
            int e = i * NTHREADS + tid;
            int d = e & 127, o = (e >> 7) * ZS + d;
            float gv = gl[o];
            if (!(fabsf(gv) <= FLT_MAX_)) { illm |= (1u << i); gv = 0.f; }
            rl[o] -= 0.5f * epsl[d] * gv;
        }
        __syncthreads();

        // ---- leapfrog interior ----
        for (int s = 1; s <= L_STEPS; ++s) {
#pragma unroll
            for (int i = 0; i < EPB; ++i) {           // drift (inv_m == 1)
                int e = i * NTHREADS + tid;
                int d = e & 127, o = (e >> 7) * ZS + d;
                zl[o] += epsl[d] * rl[o];
            }
            __syncthreads();
            if (s < L_STEPS) {
                gemm_zA(zl, Al, gl, tid);
                __syncthreads();
#pragma unroll
                for (int i = 0; i < EPB; ++i) {       // full kick
                    int e = i * NTHREADS + tid;
                    int d = e & 127, o = (e >> 7) * ZS + d;
                    float gv = gl[o];
                    if (!(fabsf(gv) <= FLT_MAX_)) { illm |= (1u << i); gv = 0.f; }
                    rl[o] -= epsl[d] * gv;
                }
                __syncthreads();
            }
        }

        // ---- final gradient + half kick ----
        gemm_zA(zl, Al, gl, tid);
        __syncthreads();
#pragma unroll
        for (int i = 0; i < EPB; ++i) {
            int e = i * NTHREADS + tid;
            int d = e & 127, o = (e >> 7) * ZS + d;
            float gv = gl[o];
            if (!(fabsf(gv) <= FLT_MAX_)) { illm |= (1u << i); gv = 0.f; }
            rl[o] -= 0.5f * epsl[d] * gv;
        }
        if (illm) *anyill = 1;   // benign same-value race
        __syncthreads();

        if (*anyill) {           // rare path: per-element revert, recompute g at reverted z
#pragma unroll
            for (int i = 0; i < EPB; ++i) {
                if ((illm >> i) & 1u) {
                    int e = i * NTHREADS + tid;
                    int o = (e >> 7) * ZS + (e & 127);
                    zl[o] = zsv[o];
                    rl[o] = noise_r[nbase + e];
                }
            }
            __syncthreads();
            gemm_zA(zl, Al, gl, tid);
            __syncthreads();
        }

        // ---- Metropolis: pot_end from reused GEMM, accept per chain ----
        if (tid < RPB) {
            float pot = 0.f, kin = 0.f;
            for (int d = 0; d < DIM; ++d) {
                pot += gl[tid * ZS + d] * zl[tid * ZS + d];
                float rv = rl[tid * ZS + d]; kin += rv * rv;
            }
            float dH = H0 - 0.5f * pot - 0.5f * kin;
            float p  = expf(dH);
            if (!(fabsf(p) <= FLT_MAX_)) p = 0.f;    // non-finite exp(dH) -> 0
            p = fminf(1.f, p);
            float u = u_accept[(size_t)t * (BATCH * CHAINS) + row0 + tid];
            accl[tid] = (u < p) ? 1.f : 0.f;
        }
        __syncthreads();

        // ---- select accepted state, emit z_stack[t+1] ----
        float* zout = z_stack + (size_t)(t + 1) * stride_t + (size_t)row0 * DIM;
#pragma unroll
        for (int i = 0; i < EPB; ++i) {
            int e = i * NTHREADS + tid;
            int rr = e >> 7, o = rr * ZS + (e & 127);
            float zv = (accl[rr] != 0.f) ? zl[o] : zsv[o];
            zl[o] = zv;
            zout[e] = zv;
        }
        __syncthreads();
    }

    // ---- z_final ----
#pragma unroll
    for (int i = 0; i < EPB; ++i) {
        int e = i * NTHREADS + tid;
        z_final[(size_t)row0 * DIM + e] = zl[(e >> 7) * ZS + (e & 127)];
    }
}

extern "C" void kernel_launch(void* const* d_in, const int* in_sizes, int n_in,
                              void* d_out, int out_size, void* d_ws, size_t ws_size,
                              hipStream_t stream) {
    const float* mu0           = (const float*)d_in[0];
    const float* var0          = (const float*)d_in[1];
    const float* log_eps       = (const float*)d_in[2];
    const float* log_inflation = (const float*)d_in[3];
    const float* A             = (const float*)d_in[4];
    const float* noise_init    = (const float*)d_in[5];
    const float* noise_r       = (const float*)d_in[6];
    const float* u_accept      = (const float*)d_in[7];
    float* out = (float*)d_out;

    const size_t smem_bytes = (size_t)SMEM_FLOATS * sizeof(float); // ~150 KB, fits 320 KB WGP LDS
    (void)hipFuncSetAttribute((const void*)hmc_persistent_kernel,
                              hipFuncAttributeMaxDynamicSharedMemorySize,
                              (int)smem_bytes);
    hmc_persistent_kernel<<<BATCH, NTHREADS, smem_bytes, stream>>>(
        mu0, var0, log_eps, log_inflation, A, noise_init, noise_r, u_accept, out);
}